// GAT_36773509988956
// MI455X (gfx1250) — compile-verified
//
#include <hip/hip_runtime.h>
#include <math.h>

#define FDIM 64

typedef __attribute__((ext_vector_type(2))) float v2f;
typedef __attribute__((ext_vector_type(8))) float v8f;

// ---------------------------------------------------------------------------
// Float atomic max via monotone integer compare trick.
// positive floats compare like ints; negative floats as uints: smaller uint
// == larger float, so unsigned min realizes max. Init value must be -inf.
// ---------------------------------------------------------------------------
__device__ __forceinline__ void atomicMaxF32(float* addr, float val) {
    if (val >= 0.0f) {
        atomicMax((int*)addr, __float_as_int(val));
    } else {
        atomicMin((unsigned int*)addr, __float_as_uint(val));
    }
}

// ---------------------------------------------------------------------------
// h = X @ W  (16 rows per wave, full 64 cols), fused hs = h@a_s, hd = h@a_d.
// Pure fp32 WMMA: V_WMMA_F32_16X16X4_F32, 16 K-steps x 4 col tiles.
//
// A-frag layout (16x4 f32, MxK): lane = M + 16*(K>=2), vgpr = K&1
//   -> lanes 0-15 hold K={0,1}, lanes 16-31 hold K={2,3}.
// B-frag layout (4x16 f32, KxN): lane = N + 16*(K>=2), vgpr = K&1.
// C/D (16x16 f32): vgpr r holds row M=r (lanes 0-15) / M=r+8 (lanes 16-31),
//   col N = lane%16.
// ---------------------------------------------------------------------------
__global__ __launch_bounds__(32) void gat_gemm_wmma(
    const float* __restrict__ X,    // [N,64]
    const float* __restrict__ W,    // [64,64] row-major
    const float* __restrict__ a_s,  // [64]
    const float* __restrict__ a_d,  // [64]
    float* __restrict__ H,          // [N,64] out
    float* __restrict__ HS,         // [N] out
    float* __restrict__ HD)         // [N] out
{
    const int lane  = threadIdx.x;       // 0..31
    const int m     = lane & 15;
    const int khalf = lane >> 4;         // 0 -> K {0,1}, 1 -> K {2,3}
    const int row0  = blockIdx.x * 16;   // N is a multiple of 16 (50000)

    // Preload A fragments for all 16 K-steps of this 16x64 tile.
    const float* xr = X + (size_t)(row0 + m) * FDIM;
    v2f af[16];
#pragma unroll
    for (int k = 0; k < 16; ++k) {
        const int kc = 4 * k + 2 * khalf;
        af[k].x = xr[kc];
        af[k].y = xr[kc + 1];
    }

    v8f acc[4];
#pragma unroll
    for (int j = 0; j < 4; ++j) acc[j] = (v8f){0,0,0,0,0,0,0,0};

#pragma unroll
    for (int j = 0; j < 4; ++j) {
#pragma unroll
        for (int k = 0; k < 16; ++k) {
            const float* wp = W + (size_t)(4 * k + 2 * khalf) * FDIM + j * 16 + m;
            v2f bf;
            bf.x = wp[0];
            bf.y = wp[FDIM];
            acc[j] = __builtin_amdgcn_wmma_f32_16x16x4_f32(
                /*neg_a=*/false, af[k], /*neg_b=*/false, bf,
                /*c_mod=*/(short)0, acc[j],
                /*reuse_a=*/false, /*reuse_b=*/false);
        }
    }

    // Write H. Lane group (khalf) owns rows row0 + 8*khalf + r, r=0..7.
    const int drow_base = row0 + 8 * khalf;
#pragma unroll
    for (int r = 0; r < 8; ++r) {
        float* hp = H + (size_t)(drow_base + r) * FDIM + m;
#pragma unroll
        for (int j = 0; j < 4; ++j) hp[j * 16] = acc[j][r];
    }

    // Fused GEMVs: hs[row] = sum_c h[row][c]*a_s[c]; same for a_d.
    float as_c[4], ad_c[4];
#pragma unroll
    for (int j = 0; j < 4; ++j) {
        as_c[j] = a_s[j * 16 + m];
        ad_c[j] = a_d[j * 16 + m];
    }
#pragma unroll
    for (int r = 0; r < 8; ++r) {
        float ps = 0.0f, pd = 0.0f;
#pragma unroll
        for (int j = 0; j < 4; ++j) {
            ps += acc[j][r] * as_c[j];
            pd += acc[j][r] * ad_c[j];
        }
        // reduce across the 16 lanes of this half-wave (xor stays in-group)
#pragma unroll
        for (int off = 8; off >= 1; off >>= 1) {
            ps += __shfl_xor(ps, off, 32);
            pd += __shfl_xor(pd, off, 32);
        }
        if (m == 0) {
            HS[drow_base + r] = ps;
            HD[drow_base + r] = pd;
        }
    }
}

// ---------------------------------------------------------------------------
// Init: OUT = 0 (N*64), segmax = -inf, segsum = 0.
// ---------------------------------------------------------------------------
__global__ void gat_init(float* __restrict__ OUT, float* __restrict__ Mx,
                         float* __restrict__ Sx, int n_nodes, int tot)
{
    int i = blockIdx.x * blockDim.x + threadIdx.x;
    if (i < tot) OUT[i] = 0.0f;
    if (i < n_nodes) {
        Mx[i] = __int_as_float(0xFF800000);  // -inf
        Sx[i] = 0.0f;
    }
}

// ---------------------------------------------------------------------------
// Per-edge raw score + leaky relu + segment max over dst.
// Edges [0,E) from edge_index; [E, Etot) are self loops (node e-E).
// ---------------------------------------------------------------------------
__global__ void gat_edge_score(const int* __restrict__ src,
                               const int* __restrict__ dst,
                               const float* __restrict__ HS,
                               const float* __restrict__ HD,
                               float* __restrict__ EW,
                               float* __restrict__ Mx,
                               int E, int Etot)
{
    int e = blockIdx.x * blockDim.x + threadIdx.x;
    if (e >= Etot) return;
    int s, d;
    if (e < E) { s = src[e]; d = dst[e]; } else { s = d = e - E; }
    float v = HS[s] + HD[d];
    v = (v >= 0.0f) ? v : 0.2f * v;   // leaky_relu, NEG_SLOPE=0.2
    EW[e] = v;
    atomicMaxF32(&Mx[d], v);
}

// ---------------------------------------------------------------------------
// w = exp(e - m[dst]); segment sum over dst.
// ---------------------------------------------------------------------------
__global__ void gat_edge_exp(const int* __restrict__ dst,
                             const float* __restrict__ Mx,
                             float* __restrict__ EW,
                             float* __restrict__ Sx,
                             int E, int Etot)
{
    int e = blockIdx.x * blockDim.x + threadIdx.x;
    if (e >= Etot) return;
    int d = (e < E) ? dst[e] : (e - E);
    float w = __expf(EW[e] - Mx[d]);
    EW[e] = w;
    atomicAdd(&Sx[d], w);
}

// ---------------------------------------------------------------------------
// out[dst] += (w/s[dst]) * h[src].  64 threads = one feature each:
// coalesced reads of h[src,*], coalesced float atomics into out[dst,*].
// ---------------------------------------------------------------------------
#define EDGES_PER_BLOCK 16
__global__ __launch_bounds__(64) void gat_aggr(
    const int* __restrict__ src, const int* __restrict__ dst,
    const float* __restrict__ EW, const float* __restrict__ Sx,
    const float* __restrict__ H, float* __restrict__ OUT,
    int E, int Etot)
{
    const int f  = threadIdx.x;                    // 0..63
    const int e0 = blockIdx.x * EDGES_PER_BLOCK;
#pragma unroll 1
    for (int i = 0; i < EDGES_PER_BLOCK; ++i) {
        int e = e0 + i;
        if (e >= Etot) return;
        int s, d;
        if (e < E) { s = src[e]; d = dst[e]; } else { s = d = e - E; }
        float alpha = EW[e] / Sx[d];
        atomicAdd(&OUT[(size_t)d * FDIM + f], alpha * H[(size_t)s * FDIM + f]);
    }
}

// ---------------------------------------------------------------------------
// out[i] += b[i % 64]
// ---------------------------------------------------------------------------
__global__ void gat_bias(float* __restrict__ OUT, const float* __restrict__ b,
                         int tot)
{
    int i = blockIdx.x * blockDim.x + threadIdx.x;
    if (i < tot) OUT[i] += b[i & (FDIM - 1)];
}

// ---------------------------------------------------------------------------
// Host side
// ---------------------------------------------------------------------------
extern "C" void kernel_launch(void* const* d_in, const int* in_sizes, int n_in,
                              void* d_out, int out_size, void* d_ws, size_t ws_size,
                              hipStream_t stream) {
    const float* x   = (const float*)d_in[0];
    const int*   ei  = (const int*)  d_in[1];
    const float* W1  = (const float*)d_in[2];
    const float* as1 = (const float*)d_in[3];
    const float* ad1 = (const float*)d_in[4];
    const float* b1  = (const float*)d_in[5];
    const float* W2  = (const float*)d_in[6];
    const float* as2 = (const float*)d_in[7];
    const float* ad2 = (const float*)d_in[8];
    const float* b2  = (const float*)d_in[9];

    const int N    = in_sizes[0] / FDIM;   // 50000
    const int E    = in_sizes[1] / 2;      // 800000
    const int Etot = E + N;                // edges + self loops
    const int* src = ei;
    const int* dst = ei + E;

    // Workspace carve-up (floats): H, X1, HS, HD, Mx, Sx, EW  (~30 MB)
    float* ws = (float*)d_ws;
    float* H  = ws;  ws += (size_t)N * FDIM;
    float* X1 = ws;  ws += (size_t)N * FDIM;
    float* HS = ws;  ws += N;
    float* HD = ws;  ws += N;
    float* Mx = ws;  ws += N;
    float* Sx = ws;  ws += N;
    float* EW = ws;  ws += Etot;

    const int tot = N * FDIM;

    auto layer = [&](const float* Xin, const float* W, const float* a_s,
                     const float* a_d, const float* b, float* OUT) {
        gat_gemm_wmma<<<N / 16, 32, 0, stream>>>(Xin, W, a_s, a_d, H, HS, HD);
        gat_init<<<(tot + 255) / 256, 256, 0, stream>>>(OUT, Mx, Sx, N, tot);
        gat_edge_score<<<(Etot + 255) / 256, 256, 0, stream>>>(src, dst, HS, HD,
                                                               EW, Mx, E, Etot);
        gat_edge_exp<<<(Etot + 255) / 256, 256, 0, stream>>>(dst, Mx, EW, Sx,
                                                             E, Etot);
        gat_aggr<<<(Etot + EDGES_PER_BLOCK - 1) / EDGES_PER_BLOCK, 64, 0,
                   stream>>>(src, dst, EW, Sx, H, OUT, E, Etot);
        gat_bias<<<(tot + 255) / 256, 256, 0, stream>>>(OUT, b, tot);
    };

    layer(x,  W1, as1, ad1, b1, X1);
    layer(X1, W2, as2, ad2, b2, (float*)d_out);
}